// GraphPropagationAttention_35304631174069
// MI455X (gfx1250) — compile-verified
//
#include <hip/hip_runtime.h>
#include <hip/hip_bf16.h>
#include <cstdint>

// GraphPropagationAttention for MI455X (gfx1250, wave32).
// All GEMM-shaped ops use V_WMMA_F32_16X16X4_F32 (fp32 WMMA: memory-bound
// problem, ~10 FLOP/byte vs 23.3 TB/s HBM, so fp32 exactness is free).
//
// Required workspace (floats):
//   node/nodeT/nodeX: 3 * 2048*256
//   qkv:              2048*768
//   attnS/resid:      2 * 16*512*512
//   ered:             4*512*512*4
//   wbuf:             2048*64
//   edgeA/edgeB:      2 * 4*512*512*64   (ping-pong for the reshape quirk)
// Total ~150M floats (~600 MB).

typedef __attribute__((ext_vector_type(2))) float v2f;
typedef __attribute__((ext_vector_type(8))) float v8f;

#define BB 4
#define NN 512
#define HH 4
#define NDIM 256
#define EDIM 64
#define HDD 64   // head dim

__device__ inline v8f wmma4(v2f a, v2f b, v8f c) {
  // D = A(16x4,f32) x B(4x16,f32) + C(16x16,f32)
  return __builtin_amdgcn_wmma_f32_16x16x4_f32(false, a, false, b, (short)0, c,
                                               false, false);
}

__device__ inline float lrelu(float x) { return x >= 0.f ? x : 0.1f * x; }

// ---------------------------------------------------------------------------
// Generic fp32 WMMA GEMM: C[M,Nc] = op(A[M,K] @ W[K,Nc] [+bias] [+addend])
// flags: 1=bias, 2=lrelu, 4=addend.  blockDim=(32,4); one wave per 16x16 tile.
// ---------------------------------------------------------------------------
__global__ void gpa_gemm16(const float* __restrict__ A, const float* __restrict__ W,
                           const float* __restrict__ bias,
                           const float* __restrict__ addend,
                           float* __restrict__ C, int M, int K, int Nc, int flags) {
  const int tileM = blockIdx.x * 4 + threadIdx.y;
  const int lane = threadIdx.x;
  const int half = lane >> 4, l16 = lane & 15;
  const long row0 = (long)tileM * 16;
  const int col = blockIdx.y * 16 + l16;
  v8f acc = {};
  const long arow = (row0 + l16) * (long)K;
  for (int k = 0; k < K; k += 4) {
    const int ka = k + 2 * half;
    v2f a, b;
    a.x = A[arow + ka];
    a.y = A[arow + ka + 1];
    b.x = W[(long)ka * Nc + col];
    b.y = W[(long)(ka + 1) * Nc + col];
    acc = wmma4(a, b, acc);
  }
  const float bv = (flags & 1) ? bias[col] : 0.f;
#pragma unroll
  for (int r = 0; r < 8; ++r) {
    const long row = row0 + r + 8 * half;
    float v = acc[r] + bv;
    if (flags & 2) v = lrelu(v);
    if (flags & 4) v += addend[row * Nc + col];
    C[row * Nc + col] = v;
  }
}

// ---------------------------------------------------------------------------
// ered[m,h] = sum_e E[m,e] * Wred[h,e]  (M = B*N*N rows; N-dim padded 4->16)
// ---------------------------------------------------------------------------
__global__ void gpa_ered(const float* __restrict__ E, const float* __restrict__ Wred,
                         float* __restrict__ out) {
  const int tileM = blockIdx.x * 4 + threadIdx.y;
  const int lane = threadIdx.x;
  const int half = lane >> 4, l16 = lane & 15;
  const long row0 = (long)tileM * 16;
  v8f acc = {};
  const long arow = (row0 + l16) * (long)EDIM;
  for (int k = 0; k < EDIM; k += 4) {
    const int ka = k + 2 * half;
    v2f a, b;
    a.x = E[arow + ka];
    a.y = E[arow + ka + 1];
    b.x = (l16 < HH) ? Wred[l16 * EDIM + ka] : 0.f;       // B[e][h]=Wred[h][e]
    b.y = (l16 < HH) ? Wred[l16 * EDIM + ka + 1] : 0.f;
    acc = wmma4(a, b, acc);
  }
  if (l16 < HH) {
#pragma unroll
    for (int r = 0; r < 8; ++r) out[(row0 + r + 8 * half) * HH + l16] = acc[r];
  }
}

// ---------------------------------------------------------------------------
// Raw QK^T: attn[bh,n,m] = sum_d q[b,h,n,d]*k[b,h,m,d]  (unscaled)
// grid(32 ntiles, 8, 16 bh), block(32,4): mt = blockIdx.y*4+threadIdx.y
// ---------------------------------------------------------------------------
__global__ void gpa_qk(const float* __restrict__ qkv, float* __restrict__ attn) {
  const int bh = blockIdx.z, b = bh >> 2, h = bh & 3;
  const int nt = blockIdx.x, mt = blockIdx.y * 4 + threadIdx.y;
  const int lane = threadIdx.x, half = lane >> 4, l16 = lane & 15;
  v8f acc = {};
  const int nrow = nt * 16 + l16, mcol = mt * 16 + l16;
  const long qb = ((long)(b * NN + nrow) * 3 + 0) * NDIM + h * HDD;
  const long kb = ((long)(b * NN + mcol) * 3 + 1) * NDIM + h * HDD;
  for (int k = 0; k < HDD; k += 4) {
    const int d = k + 2 * half;
    v2f a, bv;
    a.x = qkv[qb + d];
    a.y = qkv[qb + d + 1];
    bv.x = qkv[kb + d];
    bv.y = qkv[kb + d + 1];
    acc = wmma4(a, bv, acc);
  }
  const long base = ((long)bh * NN + nt * 16) * NN + (long)mt * 16;
#pragma unroll
  for (int r = 0; r < 8; ++r)
    attn[base + (long)(r + 8 * half) * NN + l16] = acc[r];
}

// ---------------------------------------------------------------------------
// Row softmax: score = raw*scale + ered + bred; resid=score; masked rows -> 0
// logits; softmax over m. One block (256 thr) per (b,h,n) row.
// ---------------------------------------------------------------------------
__global__ void gpa_softmax(float* __restrict__ attnS, const float* __restrict__ ered,
                            const float* __restrict__ bred,
                            const unsigned char* __restrict__ mask,
                            float* __restrict__ resid, float scale) {
  const int row = blockIdx.x;            // (b*H+h)*N + n
  const int n = row & (NN - 1);
  const int h = (row >> 9) & 3;
  const int b = row >> 11;
  const int t = threadIdx.x;
  const bool mrow = mask[b * NN + n] != 0;
  const float br = bred[h];
  const long base = (long)row * NN;
  const long eb = ((long)(b * NN + n) * NN) * HH + h;
  float s0 = attnS[base + t] * scale + ered[eb + (long)t * HH] + br;
  float s1 = attnS[base + t + 256] * scale + ered[eb + (long)(t + 256) * HH] + br;
  resid[base + t] = s0;
  resid[base + t + 256] = s1;
  const float l0 = mrow ? 0.f : s0, l1 = mrow ? 0.f : s1;
  __shared__ float red[256];
  red[t] = fmaxf(l0, l1);
  __syncthreads();
  for (int s = 128; s > 0; s >>= 1) {
    if (t < s) red[t] = fmaxf(red[t], red[t + s]);
    __syncthreads();
  }
  const float mx = red[0];
  __syncthreads();
  const float e0 = __expf(l0 - mx), e1 = __expf(l1 - mx);
  red[t] = e0 + e1;
  __syncthreads();
  for (int s = 128; s > 0; s >>= 1) {
    if (t < s) red[t] += red[t + s];
    __syncthreads();
  }
  const float inv = 1.f / red[0];
  attnS[base + t] = e0 * inv;
  attnS[base + t + 256] = e1 * inv;
}

// ---------------------------------------------------------------------------
// AV: nodeT[b,n,h*64+d] = sum_m attnS[bh,n,m] * v[b,h,m,d]
// grid(32 ntiles, 1, 16 bh), block(32,4): dt = threadIdx.y (4 d-tiles)
// ---------------------------------------------------------------------------
__global__ void gpa_av(const float* __restrict__ attnS, const float* __restrict__ qkv,
                       float* __restrict__ nodeT) {
  const int bh = blockIdx.z, b = bh >> 2, h = bh & 3;
  const int nt = blockIdx.x, dt = threadIdx.y;
  const int lane = threadIdx.x, half = lane >> 4, l16 = lane & 15;
  v8f acc = {};
  const long abase = ((long)bh * NN + nt * 16 + l16) * NN;
  const int dcol = dt * 16 + l16;
  for (int k = 0; k < NN; k += 4) {
    const int km = k + 2 * half;
    v2f a, bv;
    a.x = attnS[abase + km];
    a.y = attnS[abase + km + 1];
    bv.x = qkv[((long)(b * NN + km) * 3 + 2) * NDIM + h * HDD + dcol];
    bv.y = qkv[((long)(b * NN + km + 1) * 3 + 2) * NDIM + h * HDD + dcol];
    acc = wmma4(a, bv, acc);
  }
#pragma unroll
  for (int r = 0; r < 8; ++r) {
    const int n = nt * 16 + r + 8 * half;
    nodeT[(long)(b * NN + n) * NDIM + h * HDD + dcol] = acc[r];
  }
}

// ---------------------------------------------------------------------------
// edge_new[b,e,n,m] = sum_h (attnS+resid)[b,h,n,m]*Wexp[e,h] + bexp[e]
// written in [B,E,N,N] layout; fused online softmax over m produces
// wbuf[b,n,e] = sum_m softmax_m(masked edge)*edge.  One block per (b,n).
// ---------------------------------------------------------------------------
__global__ void gpa_expand(const float* __restrict__ attnS, const float* __restrict__ resid,
                           const float* __restrict__ Wexp, const float* __restrict__ bexp,
                           const unsigned char* __restrict__ mask,
                           float* __restrict__ edgeOut, float* __restrict__ wbuf) {
  const int n = blockIdx.x, b = blockIdx.y;
  const int t = threadIdx.x;
  __shared__ float s[HH][NN];
  for (int i = t; i < HH * NN; i += 256) {
    const int h = i >> 9, m = i & (NN - 1);
    const long idx = ((long)(b * HH + h) * NN + n) * NN + m;
    s[h][m] = attnS[idx] + resid[idx];
  }
  __syncthreads();
  const bool mrow = mask[b * NN + n] != 0;
  const int e = t >> 2, part = t & 3;
  const float w0 = Wexp[e * HH + 0], w1 = Wexp[e * HH + 1];
  const float w2 = Wexp[e * HH + 2], w3 = Wexp[e * HH + 3];
  const float be = bexp[e];
  float M = -1e30f, S = 0.f, T = 0.f;
  const long obase = ((long)(b * EDIM + e) * NN + n) * NN;
  for (int mi = 0; mi < 128; ++mi) {
    const int m = mi * 4 + part;
    const float v = be + w0 * s[0][m] + w1 * s[1][m] + w2 * s[2][m] + w3 * s[3][m];
    edgeOut[obase + m] = v;
    const float lg = mrow ? 0.f : v;
    if (lg > M) {
      const float c = __expf(M - lg);
      S *= c; T *= c; M = lg;
    }
    const float ee = __expf(lg - M);
    S += ee; T += ee * v;
  }
  __shared__ float mM[256], mS[256], mT[256];
  mM[t] = M; mS[t] = S; mT[t] = T;
  __syncthreads();
  if (part == 0) {
    float Mm = mM[t], Sm = mS[t], Tm = mT[t];
#pragma unroll
    for (int p = 1; p < 4; ++p) {
      const float M2 = mM[t + p], S2 = mS[t + p], T2 = mT[t + p];
      const float nm = fmaxf(Mm, M2);
      const float c1 = __expf(Mm - nm), c2 = __expf(M2 - nm);
      Sm = Sm * c1 + S2 * c2;
      Tm = Tm * c1 + T2 * c2;
      Mm = nm;
    }
    wbuf[(long)(b * NN + n) * EDIM + e] = Tm / Sm;
  }
}

// ---------------------------------------------------------------------------
extern "C" void kernel_launch(void* const* d_in, const int* in_sizes, int n_in,
                              void* d_out, int out_size, void* d_ws, size_t ws_size,
                              hipStream_t stream) {
  const float* node_embeds = (const float*)d_in[0];   // [4,512,256]
  const float* edge_embeds = (const float*)d_in[1];   // [4,512,512,64]
  const unsigned char* pmask = (const unsigned char*)d_in[2]; // [4,512] bool
  const float* Wn   = (const float*)d_in[3];
  const float* bn   = (const float*)d_in[4];
  const float* We   = (const float*)d_in[5];
  const float* be   = (const float*)d_in[6];
  const float* Wqkv = (const float*)d_in[7];
  const float* Wproj= (const float*)d_in[8];
  const float* bproj= (const float*)d_in[9];
  const float* Wred = (const float*)d_in[10];
  const float* bred = (const float*)d_in[11];
  const float* Wexp = (const float*)d_in[12];
  const float* bexp = (const float*)d_in[13];
  const float* Wfc  = (const float*)d_in[14];
  const float* bfc  = (const float*)d_in[15];

  const float scale = 0.125f;  // (END//H)^-0.5 = 64^-0.5

  float* ws = (float*)d_ws;
  size_t off = 0;
  const size_t MN = (size_t)BB * NN;          // 2048
  const size_t BNN = MN * NN;                 // 1,048,576
  float* node  = ws + off; off += MN * NDIM;
  float* nodeT = ws + off; off += MN * NDIM;
  float* nodeX = ws + off; off += MN * NDIM;
  float* qkv   = ws + off; off += MN * 3 * NDIM;
  float* attnS = ws + off; off += (size_t)BB * HH * NN * NN;
  float* resid = ws + off; off += (size_t)BB * HH * NN * NN;
  float* ered  = ws + off; off += BNN * HH;
  float* wbuf  = ws + off; off += MN * EDIM;
  float* edgeA = ws + off; off += BNN * EDIM;
  float* edgeB = ws + off; off += BNN * EDIM;

  const dim3 blk(32, 4);

  // node = lrelu(node_embeds @ Wn + bn)           [2048,256]x[256,256]
  gpa_gemm16<<<dim3(32, 16), blk, 0, stream>>>(node_embeds, Wn, bn, nullptr,
                                               node, 2048, 256, 256, 1 | 2);
  // edgeA = lrelu(edge_embeds @ We + be)          [1M,64]x[64,64]
  gpa_gemm16<<<dim3(16384, 4), blk, 0, stream>>>(edge_embeds, We, be, nullptr,
                                                 edgeA, (int)BNN, 64, 64, 1 | 2);

  float* edgeIn = edgeA;
  float* edgeOut = edgeB;
  for (int layer = 0; layer < 3; ++layer) {
    // qkv = node @ Wqkv                           [2048,256]x[256,768]
    gpa_gemm16<<<dim3(32, 48), blk, 0, stream>>>(node, Wqkv, nullptr, nullptr,
                                                 qkv, 2048, 256, 768, 0);
    // ered[m,h] = edge rows x Wred^T              [1M,64]x[64,4->16]
    gpa_ered<<<dim3(16384), blk, 0, stream>>>(edgeIn, Wred, ered);
    // raw QK^T
    gpa_qk<<<dim3(32, 8, 16), blk, 0, stream>>>(qkv, attnS);
    // scores, residual save, masked softmax (in-place on attnS)
    gpa_softmax<<<dim3(BB * HH * NN), 256, 0, stream>>>(attnS, ered, bred, pmask,
                                                        resid, scale);
    // nodeT = attnS @ V
    gpa_av<<<dim3(32, 1, 16), blk, 0, stream>>>(attnS, qkv, nodeT);
    // edgeOut ([B,E,N,N]) + fused w reduction
    gpa_expand<<<dim3(NN, BB), 256, 0, stream>>>(attnS, resid, Wexp, bexp, pmask,
                                                 edgeOut, wbuf);
    // nodeX = nodeT + wbuf @ Wfc + bfc            [2048,64]x[64,256]
    gpa_gemm16<<<dim3(32, 16), blk, 0, stream>>>(wbuf, Wfc, bfc, nodeT,
                                                 nodeX, 2048, 64, 256, 1 | 4);
    // node(next) / d_out = nodeX @ Wproj + bproj  [2048,256]x[256,256]
    float* dst = (layer == 2) ? (float*)d_out : node;
    gpa_gemm16<<<dim3(32, 16), blk, 0, stream>>>(nodeX, Wproj, bproj, nullptr,
                                                 dst, 2048, 256, 256, 1);
    float* tmp = edgeIn; edgeIn = edgeOut; edgeOut = tmp;
  }
}